// HarmonicMixing_23390391894546
// MI455X (gfx1250) — compile-verified
//
#include <hip/hip_runtime.h>
#include <stdint.h>

// ---------------------------------------------------------------------------
// HarmonicMixing for MI455X (gfx1250).
//
// Memory-bound (268 MB @ 23.3 TB/s ~= 11.5 us floor). Strategy:
//   - One TDM TENSOR_LOAD_TO_LDS per workgroup stages 4 rows (16 KB) of x
//     into LDS (issued by wave 0, waited with s_wait_tensorcnt).
//   - All intra-row reuse (~4.9 reads/element from the octave down-sums)
//     is served from LDS; HBM sees exactly one read + one write per element.
//   - Results written with 128-bit global stores.
// ---------------------------------------------------------------------------

#define D_MODEL        1024
#define ROWS_PER_BLOCK 4
#define THREADS        256
#define VECS_PER_THREAD ((ROWS_PER_BLOCK * D_MODEL) / (THREADS * 4))

typedef __attribute__((ext_vector_type(4))) unsigned int v4u;
typedef __attribute__((ext_vector_type(8))) int          v8i;
typedef __attribute__((ext_vector_type(4))) int          v4i;

// 6-arg TDM builtin ships with the therock-10.0 headers; 5-arg on ROCm 7.2.
#if defined(__has_include)
#if __has_include(<hip/amd_detail/amd_gfx1250_TDM.h>)
#define HM_TDM_6ARG 1
#endif
#endif

__device__ __forceinline__ float hm_sigmoid(float v) {
    return 1.0f / (1.0f + __expf(-v));
}

__global__ __launch_bounds__(THREADS)
void HarmonicMixing_kernel(const float* __restrict__ x,
                           const float* __restrict__ up_w,
                           const float* __restrict__ down_w,
                           float* __restrict__ out)
{
    __shared__ float lds[ROWS_PER_BLOCK * D_MODEL];

    const size_t block_base = (size_t)blockIdx.x * (ROWS_PER_BLOCK * D_MODEL);
    const float* src = x + block_base;

    // ---- TDM: one tensor DMA (global -> LDS) issued by wave 0 --------------
    if (threadIdx.x < 32) {
        // Flat address of __shared__ = {LDS aperture, 32-bit LDS offset};
        // low 32 bits are the D#.lds_addr the TDM wants.
        uint32_t lds_off = (uint32_t)(uintptr_t)&lds[0];
        uint64_t ga      = (uint64_t)(uintptr_t)src;

        // D# group 0 (128b): count=1 | lds_addr | global_addr[56:0] | type=2
        v4u g0 = { 1u,
                   lds_off,
                   (uint32_t)(ga & 0xFFFFFFFFu),
                   (uint32_t)((ga >> 32) & 0x01FFFFFFu) | (2u << 30) };

        // D# group 1 (256b):
        //  [17:16] data_size=2 (4B)  [79:48] tensor_dim0=1024
        //  [111:80] tensor_dim1=ROWS [127:112] tile_dim0=1024
        //  [143:128] tile_dim1=ROWS  [207:160] tensor_dim0_stride=1024
        v8i g1 = { (int)(2u << 16),
                   (int)((D_MODEL & 0xFFFFu) << 16),
                   (int)(((unsigned)D_MODEL >> 16) |
                         ((unsigned)(ROWS_PER_BLOCK & 0xFFFF) << 16)),
                   (int)(((unsigned)ROWS_PER_BLOCK >> 16) |
                         ((unsigned)(D_MODEL & 0xFFFF) << 16)),
                   (int)ROWS_PER_BLOCK,   // tile_dim1; tile_dim2 = 0
                   (int)D_MODEL,          // tensor_dim0_stride[31:0]
                   0,                     // stride hi / dim1_stride lo (unused)
                   0 };

        v4i g2 = { 0, 0, 0, 0 };          // dims 2/3 unused (tile_dim3 = 0)
        v4i g3 = { 0, 0, 0, 0 };

#ifdef HM_TDM_6ARG
        v8i g4 = { 0, 0, 0, 0, 0, 0, 0, 0 };
        __builtin_amdgcn_tensor_load_to_lds(g0, g1, g2, g3, g4, 0);
#else
        __builtin_amdgcn_tensor_load_to_lds(g0, g1, g2, g3, 0);
#endif
        __builtin_amdgcn_s_wait_tensorcnt((short)0);
    }
    __syncthreads();   // publish LDS tile to all 8 waves

    // ---- per-channel mix, all operands from LDS ----------------------------
    const float uw0 = hm_sigmoid(up_w[0]);
    const float uw1 = hm_sigmoid(up_w[1]);
    const float uw2 = hm_sigmoid(up_w[2]);
    const float dw0 = hm_sigmoid(down_w[0]);
    const float dw1 = hm_sigmoid(down_w[1]);
    const float dw2 = hm_sigmoid(down_w[2]);

    float* outp = out + block_base;

#pragma unroll
    for (int g = 0; g < VECS_PER_THREAD; ++g) {
        const int vid = threadIdx.x + g * THREADS;  // float4 index in block
        const int c0  = vid << 2;                   // flat channel in block
        const int row = c0 >> 10;                   // / D_MODEL
        const int cr  = c0 & (D_MODEL - 1);         // channel within row
        const float* L = &lds[row << 10];

        const float4 v = *(const float4*)&lds[c0];
        float rr[4] = { v.x, v.y, v.z, v.w };

#pragma unroll
        for (int k = 0; k < 4; ++k) {
            const int ch = cr + k;
            float o = rr[k];
            // up-scatter: out[i*s] += uw * x[i]
            if ((ch & 1) == 0) o += uw0 * L[ch >> 1];
            if ((ch & 3) == 0) o += uw1 * L[ch >> 2];
            if ((ch & 7) == 0) o += uw2 * L[ch >> 3];
            // down-scatter: out[i/s] += dw * x[i]  (full blocks for ch >= 1)
            if (ch >= 1 && ch < (D_MODEL >> 1)) {
                const float* p = &L[ch << 1];
                o += dw0 * (p[0] + p[1]);
            }
            if (ch >= 1 && ch < (D_MODEL >> 2)) {
                const float* p = &L[ch << 2];
                o += dw1 * ((p[0] + p[1]) + (p[2] + p[3]));
            }
            if (ch >= 1 && ch < (D_MODEL >> 3)) {
                const float* p = &L[ch << 3];
                o += dw2 * (((p[0] + p[1]) + (p[2] + p[3])) +
                            ((p[4] + p[5]) + (p[6] + p[7])));
            }
            rr[k] = o;
        }

        float4 res;
        res.x = rr[0]; res.y = rr[1]; res.z = rr[2]; res.w = rr[3];
        *(float4*)&outp[c0] = res;      // global_store_b128
    }
}

extern "C" void kernel_launch(void* const* d_in, const int* in_sizes, int n_in,
                              void* d_out, int out_size, void* d_ws, size_t ws_size,
                              hipStream_t stream) {
    (void)n_in; (void)out_size; (void)d_ws; (void)ws_size;
    const float* x  = (const float*)d_in[0];
    const float* uw = (const float*)d_in[1];
    const float* dw = (const float*)d_in[2];
    float* out      = (float*)d_out;

    const int total_rows = in_sizes[0] / D_MODEL;          // 8*4096 = 32768
    const int blocks     = total_rows / ROWS_PER_BLOCK;    // 8192

    HarmonicMixing_kernel<<<blocks, THREADS, 0, stream>>>(x, uw, dw, out);
}